// AMRBART_VocabEmbed_87316685128183
// MI455X (gfx1250) — compile-verified
//
#include <hip/hip_runtime.h>

// Problem constants (match reference).
#define V_SIZE 50265
#define D_DIM  1024
#define B_SZ   16
#define T_LEN  4096
#define G_SZ   1024
#define CHUNK  128          // tokens per main-kernel block

#define AS1 __attribute__((address_space(1)))
#define AS3 __attribute__((address_space(3)))

typedef int v4i __attribute__((ext_vector_type(4)));

// ---------------------------------------------------------------------------
// Kernel 1: per-chunk sorted-segment mean.
// Block = (b, stream, chunk of 128 tokens). 8 waves x 32 lanes; wave w owns
// D-slice [w*128, w*128+128), lane holds a float4. Stores means only for
// groups fully interior to this chunk. Boundary-crossing groups are skipped
// (fixup kernel owns them) so every output row has exactly one writer.
// ---------------------------------------------------------------------------
__global__ __launch_bounds__(256)
void amrbart_segmean_main(const float* __restrict__ embed,
                          const int* __restrict__ amr_ids,
                          const int* __restrict__ amr_segs,
                          const int* __restrict__ txt_ids,
                          const int* __restrict__ txt_segs,
                          float* __restrict__ out)
{
    __shared__ __align__(16) int s_seg[CHUNK];
    __shared__ __align__(16) int s_ids[CHUNK];

    const int nchunks = T_LEN / CHUNK;              // 32
    const int blk   = blockIdx.x;                   // [0, B*2*nchunks)
    const int chunk = blk % nchunks;
    const int bs    = blk / nchunks;                // [0, 32)
    const int s     = bs & 1;                       // 0 = amr, 1 = text
    const int b     = bs >> 1;

    const int* __restrict__ ids  = (s == 0) ? amr_ids  : txt_ids;
    const int* __restrict__ segs = (s == 0) ? amr_segs : txt_segs;

    const int c0  = chunk * CHUNK;
    const int c1  = c0 + CHUNK;
    const int tid = threadIdx.x;

    // ---- stage this chunk's seg-ids and token-ids into LDS ----------------
#if defined(__gfx1250__) && __has_builtin(__builtin_amdgcn_global_load_async_to_lds_b128)
    // CDNA5 async global->LDS copy path (ASYNCcnt). 16 bytes per lane:
    // wave0 -> 128 seg ids, wave1 -> 128 token ids.
    if (tid < 64) {
        const int* gsrc;
        int*       ldst;
        if (tid < 32) { gsrc = segs + (size_t)b * T_LEN + c0 + tid * 4;        ldst = s_seg + tid * 4; }
        else          { int u = tid - 32;
                        gsrc = ids  + (size_t)b * T_LEN + c0 + u * 4;          ldst = s_ids + u * 4; }
        __builtin_amdgcn_global_load_async_to_lds_b128((AS1 v4i*)gsrc, (AS3 v4i*)ldst, 0, 0);
    }
  #if __has_builtin(__builtin_amdgcn_s_wait_asynccnt)
    __builtin_amdgcn_s_wait_asynccnt(0);
  #else
    asm volatile("s_wait_asynccnt 0" ::: "memory");
  #endif
#else
    for (int i = tid; i < CHUNK; i += 256) {
        s_seg[i] = segs[(size_t)b * T_LEN + c0 + i];
        s_ids[i] = ids [(size_t)b * T_LEN + c0 + i];
    }
#endif
    __syncthreads();

    const int lane  = tid & 31;
    const int w     = tid >> 5;                     // 0..7 -> D-slice
    const int dbase = w * 128 + lane * 4;

    const float* __restrict__ erow = embed + dbase;
    // Groups touching a chunk boundary are owned by the fixup kernel.
    const int leftg  = (c0 > 0)     ? segs[(size_t)b * T_LEN + c0 - 1] : -1;
    const int rightg = (c1 < T_LEN) ? segs[(size_t)b * T_LEN + c1]     : -1;

    float* __restrict__ obase =
        out + ((size_t)b * (2 * G_SZ) + (size_t)s * G_SZ) * D_DIM + dbase;

    float4 acc = make_float4(0.f, 0.f, 0.f, 0.f);
    int cur = s_seg[0];
    int cnt = 0;

    for (int t = 0; t < CHUNK; ++t) {
        const int sid = s_seg[t];
        if (sid != cur) {
            if (cur != leftg && cur != rightg) {     // fully interior -> store mean
                const float inv = 1.0f / (float)cnt;
                float4 r;
                r.x = acc.x * inv; r.y = acc.y * inv;
                r.z = acc.z * inv; r.w = acc.w * inv;
                *(float4*)(obase + (size_t)cur * D_DIM) = r;
            }
            cur = sid;
            cnt = 0;
            acc = make_float4(0.f, 0.f, 0.f, 0.f);
        }
        // prefetch the gather row a few tokens ahead (global_prefetch_b8)
        const int tp = t + 8;
        if (tp < CHUNK)
            __builtin_prefetch(erow + (size_t)s_ids[tp] * D_DIM, 0, 0);

        const float4 e = *(const float4*)(erow + (size_t)s_ids[t] * D_DIM);
        acc.x += e.x; acc.y += e.y; acc.z += e.z; acc.w += e.w;
        ++cnt;
    }
    if (cur != leftg && cur != rightg) {             // flush last group
        const float inv = 1.0f / (float)cnt;
        float4 r;
        r.x = acc.x * inv; r.y = acc.y * inv;
        r.z = acc.z * inv; r.w = acc.w * inv;
        *(float4*)(obase + (size_t)cur * D_DIM) = r;
    }
}

// ---------------------------------------------------------------------------
// Kernel 2: fixup. One wave per (b, stream, g). Binary-search the sorted seg
// array for this group's token range. Empty group -> write zeros. Group that
// crosses a CHUNK boundary -> recompute its full mean here (expected ~31 such
// groups of ~4 tokens per sample; worst case still correct). Otherwise the
// main kernel already wrote the row.
// ---------------------------------------------------------------------------
__global__ __launch_bounds__(256)
void amrbart_segmean_fixup(const float* __restrict__ embed,
                           const int* __restrict__ amr_ids,
                           const int* __restrict__ amr_segs,
                           const int* __restrict__ txt_ids,
                           const int* __restrict__ txt_segs,
                           float* __restrict__ out)
{
    const int lane  = threadIdx.x & 31;
    const int wv    = threadIdx.x >> 5;
    const int gflat = blockIdx.x * 8 + wv;          // [0, B*2*G)
    const int g  = gflat & (G_SZ - 1);
    const int bs = gflat >> 10;                     // / G_SZ
    const int s  = bs & 1;                          // 0 = amr, 1 = text
    const int b  = bs >> 1;

    const int* __restrict__ ids  = ((s == 0) ? amr_ids  : txt_ids)  + (size_t)b * T_LEN;
    const int* __restrict__ segs = ((s == 0) ? amr_segs : txt_segs) + (size_t)b * T_LEN;

    // lower_bound(g)
    int lo = 0, hi = T_LEN;
    while (lo < hi) { const int m = (lo + hi) >> 1; if (segs[m] <  g) lo = m + 1; else hi = m; }
    const int first = lo;
    // upper_bound(g)
    hi = T_LEN;
    while (lo < hi) { const int m = (lo + hi) >> 1; if (segs[m] <= g) lo = m + 1; else hi = m; }
    const int last = lo;
    const int cnt  = last - first;

    float* __restrict__ orow =
        out + ((size_t)b * (2 * G_SZ) + (size_t)s * G_SZ + (size_t)g) * D_DIM + lane * 4;

    if (cnt == 0) {                                  // empty group -> zeros
        const float4 z = make_float4(0.f, 0.f, 0.f, 0.f);
        #pragma unroll
        for (int j = 0; j < 8; ++j)
            *(float4*)(orow + j * 128) = z;
        return;
    }
    if ((first / CHUNK) == ((last - 1) / CHUNK))
        return;                                      // interior: main kernel wrote it

    // Boundary-crossing group: recompute full mean (wave covers all 1024 dims).
    float4 acc[8];
    #pragma unroll
    for (int j = 0; j < 8; ++j) acc[j] = make_float4(0.f, 0.f, 0.f, 0.f);

    for (int t = first; t < last; ++t) {
        const float* __restrict__ e = embed + (size_t)ids[t] * D_DIM + lane * 4;
        #pragma unroll
        for (int j = 0; j < 8; ++j) {
            const float4 v = *(const float4*)(e + j * 128);
            acc[j].x += v.x; acc[j].y += v.y; acc[j].z += v.z; acc[j].w += v.w;
        }
    }
    const float inv = 1.0f / (float)cnt;
    #pragma unroll
    for (int j = 0; j < 8; ++j) {
        float4 r;
        r.x = acc[j].x * inv; r.y = acc[j].y * inv;
        r.z = acc[j].z * inv; r.w = acc[j].w * inv;
        *(float4*)(orow + j * 128) = r;
    }
}

// ---------------------------------------------------------------------------
extern "C" void kernel_launch(void* const* d_in, const int* in_sizes, int n_in,
                              void* d_out, int out_size, void* d_ws, size_t ws_size,
                              hipStream_t stream) {
    (void)in_sizes; (void)n_in; (void)out_size; (void)d_ws; (void)ws_size;
    const float* embed    = (const float*)d_in[0];
    const int*   txt_ids  = (const int*)  d_in[1];
    const int*   txt_segs = (const int*)  d_in[2];
    const int*   amr_ids  = (const int*)  d_in[3];
    const int*   amr_segs = (const int*)  d_in[4];
    float*       out      = (float*)      d_out;

    const int main_blocks  = B_SZ * 2 * (T_LEN / CHUNK);   // 1024
    const int fixup_blocks = (B_SZ * 2 * G_SZ) / 8;        // 4096

    amrbart_segmean_main<<<main_blocks, 256, 0, stream>>>(
        embed, amr_ids, amr_segs, txt_ids, txt_segs, out);
    amrbart_segmean_fixup<<<fixup_blocks, 256, 0, stream>>>(
        embed, amr_ids, amr_segs, txt_ids, txt_segs, out);
}